// SemiCRFDecoder_72146860638316
// MI455X (gfx1250) — compile-verified
//
#include <hip/hip_runtime.h>
#include <math.h>

// Problem constants (from reference): B=4, T=256, L=128, E=64
#define BB 4
#define TT 256
#define LL 128
#define EE 64

typedef __attribute__((ext_vector_type(2))) float v2f;
typedef __attribute__((ext_vector_type(8))) float v8f;

// -------------------------------------------------------------------------
// Kernel 1: dot[b][l][t] = sum_e harmony[b][l][e] * note[b][t][e]
// One wave32 per 16x16 tile of dot[b]; f32 WMMA, K=4 per step, 16 steps.
// A-frag (16x4 f32): lane m = lane&15; VGPR0/1 hold K = 2*(lane>>4) + {0,1}
// B-frag (4x16 f32): lane n = lane&15; VGPR0/1 hold K = 2*(lane>>4) + {0,1}
// C/D (16x16 f32):   VGPR j -> (M = j + 8*(lane>>4), N = lane&15)
// -------------------------------------------------------------------------
__global__ __launch_bounds__(32) void segscore_dot_wmma(
    const float* __restrict__ note,   // [B][T][E]
    const float* __restrict__ harm,   // [B][L][E]
    float* __restrict__ dot)          // [B][L][T]
{
    int tile  = blockIdx.x;
    int tTile = tile % (TT / 16); tile /= (TT / 16);
    int lTile = tile % (LL / 16);
    int b     = tile / (LL / 16);

    int lane = threadIdx.x;          // 0..31, full wave (EXEC all ones)
    int half = lane >> 4;            // 0 or 1
    int lo   = lane & 15;

    const float* harmB = harm + (size_t)b * LL * EE + (size_t)(lTile * 16 + lo) * EE;
    const float* noteB = note + (size_t)b * TT * EE + (size_t)(tTile * 16 + lo) * EE;

    v8f c = {};
#pragma unroll
    for (int k0 = 0; k0 < EE; k0 += 4) {
        int ka = k0 + 2 * half;
        v2f a  = { harmB[ka], harmB[ka + 1] };   // A[m][k], A[m][k+1]
        v2f bf = { noteB[ka], noteB[ka + 1] };   // B[k][n], B[k+1][n] (= note[n][k])
        // 8 args: (neg_a, A, neg_b, B, c_mod, C, reuse_a, reuse_b)
        c = __builtin_amdgcn_wmma_f32_16x16x4_f32(
                false, a, false, bf, (short)0, c, false, false);
    }

    float* dB = dot + (size_t)b * LL * TT;
    int tcol = tTile * 16 + lo;
#pragma unroll
    for (int j = 0; j < 8; ++j) {
        int lrow = lTile * 16 + j + 8 * half;
        dB[(size_t)lrow * TT + tcol] = c[j];
    }
}

// -------------------------------------------------------------------------
// Kernel 2: per (b,l) row: m = max_t dot/sqrt(E); w_t = exp(dot_t/8 - m);
// W[b][t][l] = prefix sum of w;  P[b][t][l] = prefix sum of w*dot.
// (stabilizer m cancels in the final ratio -> matches reference softmax)
// Tiny: B*L = 512 threads, serial T=256 scan each.
// -------------------------------------------------------------------------
__global__ __launch_bounds__(256) void segscore_scan(
    const float* __restrict__ dot,    // [B][L][T]
    float* __restrict__ Wp,           // [B][T][L]
    float* __restrict__ Pp)           // [B][T][L]
{
    int bl = blockIdx.x * blockDim.x + threadIdx.x;
    if (bl >= BB * LL) return;
    int b = bl / LL, l = bl % LL;

    const float scale = 0.125f;       // 1/sqrt(64)
    const float* row = dot + ((size_t)b * LL + l) * TT;

    float m = -INFINITY;
#pragma unroll 4
    for (int t = 0; t < TT; ++t) m = fmaxf(m, row[t] * scale);

    float w = 0.0f, p = 0.0f;
    for (int t = 0; t < TT; ++t) {
        float d = row[t];
        float e = __expf(fmaf(d, scale, -m));
        w += e;
        p = fmaf(e, d, p);
        Wp[((size_t)b * TT + t) * LL + l] = w;
        Pp[((size_t)b * TT + t) * LL + l] = p;
    }
}

// -------------------------------------------------------------------------
// Kernel 3 (the HBM-bound one): out[b][s][e][l] =
//   e>=s ? (P[e]-P[s-1]) / max(W[e]-W[s-1], 1e-30) : 0
// One block per (b,s). W/P rows live in L2 (1 MB total); stores are the
// bottleneck -> float4 (global_store_b128), fully coalesced over (e,l).
// -------------------------------------------------------------------------
__global__ __launch_bounds__(256) void segscore_out(
    const float* __restrict__ Wp,     // [B][T][L]
    const float* __restrict__ Pp,     // [B][T][L]
    float* __restrict__ out)          // [B][T][T][L]
{
    int b = blockIdx.x / TT;
    int s = blockIdx.x % TT;
    int tid = threadIdx.x;

    __shared__ float Ws[LL];
    __shared__ float Ps[LL];
    if (tid < LL) {
        if (s > 0) {
            Ws[tid] = Wp[((size_t)b * TT + (s - 1)) * LL + tid];
            Ps[tid] = Pp[((size_t)b * TT + (s - 1)) * LL + tid];
        } else {
            Ws[tid] = 0.0f;
            Ps[tid] = 0.0f;
        }
    }
    __syncthreads();

    const float4* W4 = (const float4*)(Wp + (size_t)b * TT * LL);
    const float4* P4 = (const float4*)(Pp + (size_t)b * TT * LL);
    float4* o4 = (float4*)(out + (((size_t)b * TT + s) * TT) * LL);

    const int QPR = LL / 4;                 // 32 float4 per e-row
    for (int q = tid; q < TT * QPR; q += 256) {
        int e  = q / QPR;                   // end index
        int l4 = (q % QPR) * 4;             // first l of this quad
        float4 r;
        if (e >= s) {
            float4 we = W4[q];
            float4 pe = P4[q];
            r.x = (pe.x - Ps[l4 + 0]) / fmaxf(we.x - Ws[l4 + 0], 1e-30f);
            r.y = (pe.y - Ps[l4 + 1]) / fmaxf(we.y - Ws[l4 + 1], 1e-30f);
            r.z = (pe.z - Ps[l4 + 2]) / fmaxf(we.z - Ws[l4 + 2], 1e-30f);
            r.w = (pe.w - Ps[l4 + 3]) / fmaxf(we.w - Ws[l4 + 3], 1e-30f);
        } else {
            r.x = r.y = r.z = r.w = 0.0f;
        }
        o4[q] = r;
    }
}

// -------------------------------------------------------------------------
extern "C" void kernel_launch(void* const* d_in, const int* in_sizes, int n_in,
                              void* d_out, int out_size, void* d_ws, size_t ws_size,
                              hipStream_t stream) {
    (void)in_sizes; (void)n_in; (void)out_size; (void)ws_size;
    const float* note = (const float*)d_in[0];   // [B,T,E]
    const float* harm = (const float*)d_in[1];   // [B,L,E]
    float* out = (float*)d_out;                  // [B,T,T,L]

    float* dot = (float*)d_ws;                                   // B*L*T  floats
    float* Wp  = dot + (size_t)BB * LL * TT;                     // B*T*L  floats
    float* Pp  = Wp  + (size_t)BB * TT * LL;                     // B*T*L  floats

    // 1) GEMM via f32 WMMA: 512 tiles, one wave32 each
    segscore_dot_wmma<<<BB * (LL / 16) * (TT / 16), 32, 0, stream>>>(note, harm, dot);
    // 2) per-row max + prefix sums (tiny)
    segscore_scan<<<(BB * LL + 255) / 256, 256, 0, stream>>>(dot, Wp, Pp);
    // 3) 128 MB coalesced output stream
    segscore_out<<<BB * TT, 256, 0, stream>>>(Wp, Pp, out);
}